// snn_input_36129264894779
// MI455X (gfx1250) — compile-verified
//
#include <hip/hip_runtime.h>
#include <hip/hip_bf16.h>
#include <stdint.h>

// Integrate-and-fire SNN scan, bandwidth-bound (~1.07 GiB traffic -> ~46us
// floor at 23.3 TB/s). CDNA5 path used: GLOBAL_LOAD_ASYNC_TO_LDS_B128 with a
// 4-deep per-wave pipeline + s_wait_asynccnt, B128 vector IO, NT stores.
// Main loop unrolled x4 so the LDS ring slot is a compile-time constant and
// all addresses are pointer-bumps (no per-iteration 64-bit multiplies).

#define TSTEPS_FAST 32
#define BLOCK 256
#define VEC 4
#define NPB (BLOCK * VEC)   // neurons per block = 1024
#define DEPTH 4             // async pipeline depth (ring buffers in LDS)
#define THRES 1.0f

typedef __attribute__((ext_vector_type(4))) float v4f;
typedef __attribute__((ext_vector_type(4))) int   v4i;

#define AS_GLOBAL __attribute__((address_space(1)))
#define AS_LDS    __attribute__((address_space(3)))

static_assert(DEPTH == 4, "pipeline code below is written for DEPTH == 4");

// ---- CDNA5 async global->LDS copy (16 bytes per lane) ----------------------
__device__ __forceinline__ void async_copy_b128(const float* gsrc, float* ldst) {
#if __has_builtin(__builtin_amdgcn_global_load_async_to_lds_b128)
  // Builtin is element-typed: b128 takes pointers to int __vector(4).
  __builtin_amdgcn_global_load_async_to_lds_b128(
      (AS_GLOBAL v4i*)gsrc, (AS_LDS v4i*)ldst, /*offset=*/0, /*cpol=*/0);
#else
  unsigned lofs = (unsigned)(size_t)(AS_LDS void*)ldst;
  asm volatile("global_load_async_to_lds_b128 %0, %1, off"
               :
               : "v"(lofs), "v"(gsrc)
               : "memory");
#endif
}

template <int K>
__device__ __forceinline__ void wait_asynccnt() {
#if __has_builtin(__builtin_amdgcn_s_wait_asynccnt)
  __builtin_amdgcn_s_wait_asynccnt(K);
#else
  asm volatile("s_wait_asynccnt %0" : : "i"(K) : "memory");
#endif
}

// ---- fast path: T == 32, N % 1024 == 0 -------------------------------------
__global__ __launch_bounds__(BLOCK) void snn_if_async(
    const float* __restrict__ image,      // [T, N]
    const float* __restrict__ mempot_in,  // [N]
    float* __restrict__ spikes,           // [T, N] (0.0/1.0)
    float* __restrict__ mempot_out,       // [N]
    int N) {
  __shared__ __align__(16) float lds[DEPTH][NPB];

  const int tid = threadIdx.x;
  const long long n4 = (long long)blockIdx.x * NPB + (long long)tid * VEC;
  const long long stride = (long long)N;  // floats between consecutive t's
  float* lbase = &lds[0][tid * VEC];      // this thread's 16B slot per depth

  v4f m = *(const v4f*)(mempot_in + n4);

  // Prologue: fill the 4-deep async pipeline (t = 0..3).
  {
    const float* g = image + n4;
#pragma unroll
    for (int d = 0; d < DEPTH; ++d) {
      async_copy_b128(g, lbase + d * NPB);
      g += stride;
    }
  }

  auto consume = [&](int slot, float* sout) {
    // Each lane reads exactly the 16 bytes its own wave's async op wrote, and
    // ASYNCcnt completes in order -> no workgroup barrier needed.
    v4f img = *(const v4f*)(lbase + slot * NPB);
    v4f spk;
#pragma unroll
    for (int c = 0; c < VEC; ++c) {
      float mm = m[c] + img[c];
      bool s = mm >= THRES;
      spk[c] = s ? 1.0f : 0.0f;
      m[c] = s ? (mm - THRES) : mm;
    }
    // Streaming output: never re-read, 512 MiB total -> keep it out of caches.
    __builtin_nontemporal_store(spk, (v4f*)sout);
  };

  const float* gnext = image + n4 + (long long)DEPTH * stride;  // src for t+4
  float* sout = spikes + n4;                                    // dst for t

  // 28 steady-state iterations; unroll x4 makes (t & 3) a literal slot index
  // and keeps the asynccnt immediate a compile-time constant.
#pragma unroll 4
  for (int t = 0; t < TSTEPS_FAST - DEPTH; ++t) {
    wait_asynccnt<DEPTH - 1>();  // load t complete; t+1..t+3 still in flight
    consume(t & (DEPTH - 1), sout);
    sout += stride;
    // Keep the LDS read of this slot ordered before the async re-fill of the
    // same slot (value consumption above already forced the dscnt wait).
    asm volatile("" ::: "memory");
    async_copy_b128(gnext, lbase + (t & (DEPTH - 1)) * NPB);
    gnext += stride;
  }

  // Drain: wait counts must be compile-time immediates.
  wait_asynccnt<3>(); consume(0, sout); sout += stride;
  wait_asynccnt<2>(); consume(1, sout); sout += stride;
  wait_asynccnt<1>(); consume(2, sout); sout += stride;
  wait_asynccnt<0>(); consume(3, sout);

  __builtin_nontemporal_store(m, (v4f*)(mempot_out + n4));
}

// ---- generic fallback (any T, any N) ---------------------------------------
__global__ void snn_if_generic(const float* __restrict__ image,
                               const float* __restrict__ mempot_in,
                               float* __restrict__ spikes,
                               float* __restrict__ mempot_out,
                               int T, int N) {
  long long n = (long long)blockIdx.x * blockDim.x + threadIdx.x;
  if (n >= N) return;
  float m = mempot_in[n];
  for (int t = 0; t < T; ++t) {
    m += image[(long long)t * N + n];
    bool s = m >= THRES;
    spikes[(long long)t * N + n] = s ? 1.0f : 0.0f;
    if (s) m -= THRES;
  }
  mempot_out[n] = m;
}

extern "C" void kernel_launch(void* const* d_in, const int* in_sizes, int n_in,
                              void* d_out, int out_size, void* d_ws, size_t ws_size,
                              hipStream_t stream) {
  const float* image  = (const float*)d_in[0];   // [T, N] f32
  const float* mempot = (const float*)d_in[1];   // [N] f32
  const long long total = (long long)in_sizes[0];
  const int N = in_sizes[1];
  const int T = (int)(total / (long long)N);

  float* spikes     = (float*)d_out;                          // [T, N]
  float* mempot_out = (float*)d_out + (size_t)T * (size_t)N;  // [N]

  if (T == TSTEPS_FAST && (N % NPB) == 0) {
    int blocks = N / NPB;
    snn_if_async<<<blocks, BLOCK, 0, stream>>>(image, mempot, spikes,
                                               mempot_out, N);
  } else {
    int blocks = (int)((N + BLOCK - 1) / BLOCK);
    snn_if_generic<<<blocks, BLOCK, 0, stream>>>(image, mempot, spikes,
                                                 mempot_out, T, N);
  }
}